// HeteroRGCNLayer_20959440404561
// MI455X (gfx1250) — compile-verified
//
#include <hip/hip_runtime.h>

typedef float v2f __attribute__((ext_vector_type(2)));
typedef float v8f __attribute__((ext_vector_type(8)));

#define IN_SIZE  128
#define OUT_SIZE 64

// ---------------------------------------------------------------------------
// Wh = X @ W + bias   via V_WMMA_F32_16X16X4_F32
// One wave -> 16 rows x 64 cols. Block = 128 threads = 4 waves = 64 rows.
// ---------------------------------------------------------------------------
__global__ __launch_bounds__(128)
void gemm_bias_wmma(const float* __restrict__ X, const float* __restrict__ W,
                    const float* __restrict__ bias, float* __restrict__ Y, int M)
{
    const int wave = threadIdx.x >> 5;
    const int lane = threadIdx.x & 31;
    const int half = lane >> 4;   // 0: K=0,1 / M=j ; 1: K=2,3 / M=j+8
    const int lrow = lane & 15;
    const int rowBase = (blockIdx.x * 4 + wave) * 16;
    if (rowBase >= M) return;     // wave-uniform exit; EXEC all-1s below

    // C/D accumulators, initialized with per-column bias (column = t*16+lrow,
    // identical for all 8 rows held by this lane).
    v8f acc[4];
    #pragma unroll
    for (int t = 0; t < 4; ++t) {
        const float bv = bias[t * 16 + lrow];
        #pragma unroll
        for (int j = 0; j < 8; ++j) acc[t][j] = bv;
    }

    // A-fragment source row for this lane (clamped for non-multiple-of-16 M;
    // duplicate loads are harmless, stores are guarded).
    int rl = rowBase + lrow; if (rl > M - 1) rl = M - 1;
    const float* xrow = X + (size_t)rl * IN_SIZE + half * 2;  // 8B aligned

    #pragma unroll 4
    for (int k0 = 0; k0 < IN_SIZE; k0 += 4) {
        // A 16x4: lane holds {K=k0+2*half, K=k0+2*half+1} of row lrow
        v2f a = *(const v2f*)(xrow + k0);
        const float* wk = W + (size_t)(k0 + half * 2) * OUT_SIZE + lrow;
        #pragma unroll
        for (int t = 0; t < 4; ++t) {
            // B 4x16: VGPR0 = row K=k0+2*half, VGPR1 = row K=k0+2*half+1,
            // column = t*16 + lrow
            v2f b;
            b.x = wk[t * 16];
            b.y = wk[t * 16 + OUT_SIZE];
            acc[t] = __builtin_amdgcn_wmma_f32_16x16x4_f32(
                /*neg_a=*/false, a, /*neg_b=*/false, b,
                /*c_mod=*/(short)0, acc[t], /*reuse_a=*/false, /*reuse_b=*/false);
        }
    }

    // C/D layout: VGPR j -> row rowBase + j + 8*half, col t*16 + lrow
    #pragma unroll
    for (int t = 0; t < 4; ++t) {
        #pragma unroll
        for (int j = 0; j < 8; ++j) {
            const int r = rowBase + half * 8 + j;
            if (r < M)
                Y[(size_t)r * OUT_SIZE + (t * 16 + lrow)] = acc[t][j];
        }
    }
}

// ---------------------------------------------------------------------------
// Per-destination degree (float) via atomics.
// ---------------------------------------------------------------------------
__global__ void edge_count_kernel(const int* __restrict__ dst,
                                  float* __restrict__ cnt, int nE)
{
    const int e = blockIdx.x * blockDim.x + threadIdx.x;
    if (e < nE) atomicAdd(&cnt[dst[e]], 1.0f);
}

// ---------------------------------------------------------------------------
// out[dst] += Wh[src] / max(cnt[dst],1)  -- one thread per (edge, column).
// Gathers are lane-contiguous (128B per wave); all hot data is L2-resident.
// ---------------------------------------------------------------------------
__global__ __launch_bounds__(256)
void edge_scatter_kernel(const float* __restrict__ Wh,
                         const int* __restrict__ src,
                         const int* __restrict__ dst,
                         const float* __restrict__ cnt,
                         float* __restrict__ out, int nE)
{
    const long long gid = (long long)blockIdx.x * blockDim.x + threadIdx.x;
    if (gid >= (long long)nE * OUT_SIZE) return;
    const int e = (int)(gid >> 6);
    const int c = (int)(gid & (OUT_SIZE - 1));
    const int s = src[e];
    const int d = dst[e];
    const float scale = 1.0f / fmaxf(cnt[d], 1.0f);
    atomicAdd(out + (size_t)d * OUT_SIZE + c,
              Wh[(size_t)s * OUT_SIZE + c] * scale);
}

// ---------------------------------------------------------------------------
// Host-side launcher (graph-capture safe: kernels + hipMemsetAsync only).
// ---------------------------------------------------------------------------
extern "C" void kernel_launch(void* const* d_in, const int* in_sizes, int n_in,
                              void* d_out, int out_size, void* d_ws, size_t ws_size,
                              hipStream_t stream)
{
    const float* user_feat = (const float*)d_in[0];
    const float* item_feat = (const float*)d_in[1];
    const float* W_clicks  = (const float*)d_in[2];
    const float* b_clicks  = (const float*)d_in[3];
    const float* W_bought  = (const float*)d_in[4];
    const float* b_bought  = (const float*)d_in[5];
    const int*   e0_src    = (const int*)d_in[6];
    const int*   e0_dst    = (const int*)d_in[7];
    const int*   e1_src    = (const int*)d_in[8];
    const int*   e1_dst    = (const int*)d_in[9];

    const int n_nodes = in_sizes[0] / IN_SIZE;   // == num_target
    const int nE0     = in_sizes[6];
    const int nE1     = in_sizes[8];

    float* out  = (float*)d_out;
    float* cnt0 = (float*)d_ws;
    float* cnt1 = cnt0 + n_nodes;
    float* Wh   = cnt1 + n_nodes;                // n_nodes*64 floats, reused per relation

    hipMemsetAsync(out,  0, (size_t)out_size * sizeof(float), stream);
    hipMemsetAsync(cnt0, 0, (size_t)2 * n_nodes * sizeof(float), stream);

    const int tb = 256;
    edge_count_kernel<<<(nE0 + tb - 1) / tb, tb, 0, stream>>>(e0_dst, cnt0, nE0);
    edge_count_kernel<<<(nE1 + tb - 1) / tb, tb, 0, stream>>>(e1_dst, cnt1, nE1);

    const int gblocks = (n_nodes + 63) / 64;

    // relation 0: user --clicks--> target
    gemm_bias_wmma<<<gblocks, 128, 0, stream>>>(user_feat, W_clicks, b_clicks, Wh, n_nodes);
    {
        const long long total = (long long)nE0 * OUT_SIZE;
        edge_scatter_kernel<<<(int)((total + tb - 1) / tb), tb, 0, stream>>>(
            Wh, e0_src, e0_dst, cnt0, out, nE0);
    }

    // relation 1: item --bought_by--> target (reuses Wh scratch; stream-ordered)
    gemm_bias_wmma<<<gblocks, 128, 0, stream>>>(item_feat, W_bought, b_bought, Wh, n_nodes);
    {
        const long long total = (long long)nE1 * OUT_SIZE;
        edge_scatter_kernel<<<(int)((total + tb - 1) / tb), tb, 0, stream>>>(
            Wh, e1_src, e1_dst, cnt1, out, nE1);
    }
}